// IoUAssigner_11244224381515
// MI455X (gfx1250) — compile-verified
//
#include <hip/hip_runtime.h>
#include <stdint.h>

#define BS          16
#define G           64
#define A_ANCH      33600
#define NUM_CLASSES 80
#define BG_IDX      80
#define THREADS     256
#define NBLK        ((A_ANCH + THREADS - 1) / THREADS)   // 132
#define IOU_THR     0.3f
#define EPS_F       1e-9f

// d_out layout (floats), N = BS*A_ANCH = 537600:
//   [0,N)        target_labels
//   [N,5N)       target_bboxes
//   [5N,85N)     target_scores
//   [85N,86N)    fg_mask
//   [86N,87N)    target_gt_idx
#define N_TOT   ((size_t)BS * A_ANCH)
#define OFF_LAB ((size_t)0)
#define OFF_BOX (N_TOT)
#define OFF_SC  (N_TOT * 5)
#define OFF_FG  (N_TOT * 85)
#define OFF_IDX (N_TOT * 86)

__device__ __forceinline__ void top3_insert(float v, float& t0, float& t1, float& t2) {
    if (v > t0)      { t2 = t1; t1 = t0; t0 = v; }
    else if (v > t1) { t2 = t1; t1 = v; }
    else if (v > t2) { t2 = v; }
}

__global__ void k_init(int* __restrict__ fg_flag) {
    if (threadIdx.x < BS) fg_flag[threadIdx.x] = 0;
}

__global__ __launch_bounds__(THREADS) void k_pass1(
    const float* __restrict__ pd_bboxes,   // (BS, A, 4)
    const float* __restrict__ gt_bboxes,   // (BS, G, 4)
    const int*   __restrict__ mask_gt,     // (BS, G, 1)
    float*       __restrict__ anchor_max,  // (BS, A)  max over valid g (-inf if none)
    int*         __restrict__ anchor_idx,  // (BS, A)  argmax of multiply-masked overlaps
    float*       __restrict__ blk_top3,    // (BS, NBLK, 3)
    int*         __restrict__ fg_flag)     // (BS)
{
    __shared__ float s_gt[G * 4];
    __shared__ int   s_val[G];
    __shared__ float s_t3[THREADS][3];
    __shared__ int   s_any;

    const int tid = threadIdx.x;
    const int b   = blockIdx.y;
    const int a0  = blockIdx.x * THREADS;
    const int a   = a0 + tid;

    if (tid == 0) s_any = 0;

    // Overlap the per-thread pd box fetch with GT staging.
    if (a < A_ANCH)
        __builtin_prefetch(pd_bboxes + ((size_t)b * A_ANCH + a) * 4, 0, 0);

    // CDNA5 async tensor path: stage the 64 GT boxes (256 dwords, one per lane)
    // straight from HBM/L2 into LDS, bypassing VGPRs. Tracked by ASYNCcnt.
    {
        uint64_t gaddr = (uint64_t)(const void*)(gt_bboxes + (size_t)b * G * 4 + tid);
        uint32_t laddr = (uint32_t)(uintptr_t)(&s_gt[tid]);  // low 32 bits = LDS offset
        asm volatile("global_load_async_to_lds_b32 %0, %1, off"
                     :: "v"(laddr), "v"(gaddr) : "memory");
    }
    if (tid < G) s_val[tid] = mask_gt[(size_t)b * G + tid];
    asm volatile("s_wait_asynccnt 0" ::: "memory");
    __syncthreads();

    float t0 = -INFINITY, t1 = -INFINITY, t2 = -INFINITY;
    bool pos = false;

    if (a < A_ANCH) {
        const float4 p = *(const float4*)(pd_bboxes + ((size_t)b * A_ANCH + a) * 4);
        const float pa = (p.z - p.x) * (p.w - p.y);
        float amax = -1.0f;      // multiply-masked values are >= 0, so first g always wins init
        int   aidx = 0;
        float m    = -INFINITY;  // max over valid g only
        #pragma unroll 4
        for (int g = 0; g < G; ++g) {
            const float gx0 = s_gt[g * 4 + 0], gy0 = s_gt[g * 4 + 1];
            const float gx1 = s_gt[g * 4 + 2], gy1 = s_gt[g * 4 + 3];
            float iw = fminf(gx1, p.z) - fmaxf(gx0, p.x); iw = fmaxf(iw, 0.0f);
            float ih = fminf(gy1, p.w) - fmaxf(gy0, p.y); ih = fmaxf(ih, 0.0f);
            const float inter = iw * ih;
            const float ga    = (gx1 - gx0) * (gy1 - gy0);
            const float iou   = inter / (ga + pa - inter + EPS_F);
            const bool  v     = (s_val[g] != 0);
            const float mv    = v ? iou : 0.0f;       // reference: overlaps * valid
            if (mv > amax) { amax = mv; aidx = g; }    // first-max tie-break (jnp.argmax)
            const float mi    = v ? iou : -INFINITY;  // reference: where(valid, ov, -inf)
            m = fmaxf(m, mi);
            top3_insert(mi, t0, t1, t2);
            pos |= (mv > IOU_THR);
        }
        anchor_max[(size_t)b * A_ANCH + a] = m;
        anchor_idx[(size_t)b * A_ANCH + a] = aidx;
    }

    s_t3[tid][0] = t0; s_t3[tid][1] = t1; s_t3[tid][2] = t2;
    if (pos) atomicOr(&s_any, 1);      // LDS atomic, cheap
    __syncthreads();

    // tree-merge sorted triples -> block top-3
    for (int s = THREADS / 2; s > 0; s >>= 1) {
        if (tid < s) {
            float u0 = s_t3[tid][0], u1 = s_t3[tid][1], u2 = s_t3[tid][2];
            #pragma unroll
            for (int k = 0; k < 3; ++k) top3_insert(s_t3[tid + s][k], u0, u1, u2);
            s_t3[tid][0] = u0; s_t3[tid][1] = u1; s_t3[tid][2] = u2;
        }
        __syncthreads();
    }

    if (tid == 0) {
        float* o = &blk_top3[((size_t)b * NBLK + blockIdx.x) * 3];
        o[0] = s_t3[0][0]; o[1] = s_t3[0][1]; o[2] = s_t3[0][2];
        if (s_any) atomicOr(&fg_flag[b], 1);   // one global atomic per block
    }
}

__global__ void k_pass2(const float* __restrict__ blk_top3,
                        const int*   __restrict__ fg_flag,
                        const int*   __restrict__ mask_gt,
                        float*       __restrict__ min_iou,
                        int*         __restrict__ need_fb)
{
    const int b = blockIdx.x;
    if (threadIdx.x != 0) return;
    float u0 = -INFINITY, u1 = -INFINITY, u2 = -INFINITY;
    const float* src = blk_top3 + (size_t)b * NBLK * 3;
    for (int i = 0; i < NBLK * 3; ++i) top3_insert(src[i], u0, u1, u2);
    int hv = 0;
    for (int g = 0; g < G; ++g) hv |= (mask_gt[(size_t)b * G + g] != 0);
    min_iou[b] = u2;
    need_fb[b] = (fg_flag[b] == 0) && hv;
}

__global__ __launch_bounds__(THREADS) void k_pass3(
    const int*   __restrict__ gt_labels,   // (BS, G, 1)
    const float* __restrict__ gt_bboxes,   // (BS, G, 4)
    const int*   __restrict__ mask_gt,     // (BS, G, 1)
    const float* __restrict__ anchor_max,
    const int*   __restrict__ anchor_idx,
    const float* __restrict__ min_iou,
    const int*   __restrict__ need_fb,
    float*       __restrict__ out)
{
    __shared__ int s_lab[THREADS];
    __shared__ int s_mk[THREADS];

    const int tid = threadIdx.x;
    const int b   = blockIdx.y;
    const int a0  = blockIdx.x * THREADS;
    const int a   = a0 + tid;

    int mk = 0, lab = 0;
    if (a < A_ANCH) {
        const size_t pa  = (size_t)b * A_ANCH + a;
        const float  m   = anchor_max[pa];
        const int    idx = anchor_idx[pa];
        const bool   fg  = need_fb[b] ? (m >= min_iou[b]) : (m > IOU_THR);
        const bool   va  = (mask_gt[(size_t)b * G + idx] != 0);
        mk  = (fg && va) ? 1 : 0;
        lab = gt_labels[(size_t)b * G + idx];

        out[OFF_LAB + pa] = (float)(mk ? lab : BG_IDX);
        const float4 gb = *(const float4*)(gt_bboxes + ((size_t)b * G + idx) * 4);
        const float  fm = mk ? 1.0f : 0.0f;
        float4 ob; ob.x = gb.x * fm; ob.y = gb.y * fm; ob.z = gb.z * fm; ob.w = gb.w * fm;
        *(float4*)(out + OFF_BOX + pa * 4) = ob;
        out[OFF_FG  + pa] = fg ? 1.0f : 0.0f;
        out[OFF_IDX + pa] = (float)idx;
    }
    s_lab[tid] = lab;
    s_mk[tid]  = mk;
    __syncthreads();

    // Coalesced float4 one-hot score writes for this block's 256x80 region.
    const int    cnt  = min(THREADS, A_ANCH - a0);
    const int    tot4 = cnt * NUM_CLASSES / 4;
    float* __restrict__ sb = out + OFF_SC + ((size_t)b * A_ANCH + a0) * NUM_CLASSES;
    for (int j = tid; j < tot4; j += THREADS) {
        const int e   = j * 4;
        const int anc = e / NUM_CLASSES;
        const int c   = e - anc * NUM_CLASSES;   // 80 % 4 == 0: float4 stays in one row
        const int mm  = s_mk[anc];
        const int lb  = s_lab[anc];
        float4 v;
        v.x = (mm && lb == c    ) ? 1.0f : 0.0f;
        v.y = (mm && lb == c + 1) ? 1.0f : 0.0f;
        v.z = (mm && lb == c + 2) ? 1.0f : 0.0f;
        v.w = (mm && lb == c + 3) ? 1.0f : 0.0f;
        *(float4*)(sb + e) = v;
    }
}

extern "C" void kernel_launch(void* const* d_in, const int* in_sizes, int n_in,
                              void* d_out, int out_size, void* d_ws, size_t ws_size,
                              hipStream_t stream) {
    // setup_inputs order: pd_scores, pd_bboxes, anc_points, gt_labels, gt_bboxes, mask_gt
    const float* pd_bboxes = (const float*)d_in[1];
    const int*   gt_labels = (const int*)d_in[3];
    const float* gt_bboxes = (const float*)d_in[4];
    const int*   mask_gt   = (const int*)d_in[5];
    float*       out       = (float*)d_out;

    char* w = (char*)d_ws;
    float* anchor_max = (float*)w;  w += sizeof(float) * N_TOT;
    int*   anchor_idx = (int*)w;    w += sizeof(int)   * N_TOT;
    float* blk_top3   = (float*)w;  w += sizeof(float) * BS * NBLK * 3;
    int*   fg_flag    = (int*)w;    w += sizeof(int)   * BS;
    float* min_iou_p  = (float*)w;  w += sizeof(float) * BS;
    int*   need_fb_p  = (int*)w;    w += sizeof(int)   * BS;

    k_init<<<1, 32, 0, stream>>>(fg_flag);

    dim3 grid(NBLK, BS);
    k_pass1<<<grid, THREADS, 0, stream>>>(pd_bboxes, gt_bboxes, mask_gt,
                                          anchor_max, anchor_idx, blk_top3, fg_flag);
    k_pass2<<<BS, 32, 0, stream>>>(blk_top3, fg_flag, mask_gt, min_iou_p, need_fb_p);
    k_pass3<<<grid, THREADS, 0, stream>>>(gt_labels, gt_bboxes, mask_gt,
                                          anchor_max, anchor_idx, min_iou_p, need_fb_p, out);
}